// PathPreservingNetwork_69741678952838
// MI455X (gfx1250) — compile-verified
//
#include <hip/hip_runtime.h>
#include <hip/hip_bf16.h>
#include <math.h>

// ---------------------------------------------------------------------------
// PathPreservingNetwork on MI455X (gfx1250), fp32 WMMA (V_WMMA_F32_16X16X4_F32)
//
// Layers are GEMMs with rows = (batch, path). Branch expansion: row r of the
// GEMM result produces output rows 3r (-t), 3r+1 (0), 3r+2 (+t).
// Pruning = exact top-128 by squared L2 norm, ties -> lower index (matches
// jax.lax.top_k; +/- branch pairs tie bit-exactly since (-x)^2 == x^2).
//
// GEMM blocking: one wave computes a 64(M) x 64(N) C block:
//   4 M-tiles x 4 N-tiles = 16 f32 WMMA accumulators (128 VGPRs).
//   Per K=4 step: 4 x b64 A loads + 8 x b32 B loads feed 16 WMMAs
//   (0.75 loads per wmma vs 2.25 in the unblocked version).
// ---------------------------------------------------------------------------

typedef __attribute__((ext_vector_type(2))) float v2f;
typedef __attribute__((ext_vector_type(8))) float v8f;

#define MT 4   // M-tiles per wave (64 rows)
#define NT 4   // N-tiles per wave (64 cols)

template<bool APPLY_TANH>
__global__ __launch_bounds__(256)
void layer_wmma_kernel(const float* __restrict__ A, const float* __restrict__ W,
                       const float* __restrict__ bias, float* __restrict__ out,
                       int M, int N, int K) {
    const int wid    = (blockIdx.x * blockDim.x + threadIdx.x) >> 5;
    const int lane   = threadIdx.x & 31;
    const int tilesN = N >> 6;                 // 64-wide N blocks
    const int tm     = wid / tilesN;           // 64-row M block index
    const int tn     = wid % tilesN;
    if (tm * 64 >= M) return;                  // uniform per wave: EXEC stays all-1s

    const int l15  = lane & 15;
    const int half = lane >> 4;                // 0: K+{0,1}, 1: K+{2,3}
    const int ncol = tn * 64 + l15;            // base column of this lane in B/C

    // fp32 A fragment (ISA 7.12.2): lanes 0-15 hold K=k0+{0,1} in VGPR{0,1},
    // lanes 16-31 hold K=k0+{2,3}; row = M-index = lane&15 within the 16-tile.
    const float* __restrict__ Ap[MT];
    #pragma unroll
    for (int i = 0; i < MT; ++i)
        Ap[i] = A + (size_t)(tm * 64 + i * 16 + l15) * K + (half << 1);
    const float* __restrict__ Wp = W + ncol;

    v8f acc[MT][NT];
    #pragma unroll
    for (int i = 0; i < MT; ++i)
        #pragma unroll
        for (int j = 0; j < NT; ++j)
            acc[i][j] = (v8f){};

    for (int k = 0; k < K; k += 4) {
        v2f a[MT];
        #pragma unroll
        for (int i = 0; i < MT; ++i)
            a[i] = *(const v2f*)(Ap[i] + k);

        // B fragment: VGPR0 = row (k + 2*half), VGPR1 = row (k + 2*half + 1).
        const float* __restrict__ wk = Wp + (size_t)(k + (half << 1)) * N;
        v2f b[NT];
        #pragma unroll
        for (int j = 0; j < NT; ++j) {
            b[j][0] = wk[j * 16];
            b[j][1] = wk[(size_t)N + j * 16];
        }

        #pragma unroll
        for (int i = 0; i < MT; ++i)
            #pragma unroll
            for (int j = 0; j < NT; ++j)
                acc[i][j] = __builtin_amdgcn_wmma_f32_16x16x4_f32(
                    false, a[i], false, b[j], (short)0, acc[i][j], false, false);
    }

    // Epilogue: bias + optional tanh, then expand row r -> rows 3r,3r+1,3r+2.
    #pragma unroll
    for (int i = 0; i < MT; ++i) {
        #pragma unroll
        for (int v = 0; v < 8; ++v) {
            const int    m  = i * 16 + v + (half << 3);
            const size_t r3 = (size_t)(tm * 64 + m) * 3;   // = 3*r
            #pragma unroll
            for (int j = 0; j < NT; ++j) {
                const int   o  = tn * 64 + j * 16 + l15;
                const float c  = acc[i][j][v] + bias[o];
                const float tv = APPLY_TANH ? tanhf(c) : c;
                const size_t base = r3 * (size_t)N + o;
                out[base]                 = -tv;   // branch t=0
                out[base + (size_t)N]     = 0.0f;  // branch t=1 (zero branch)
                out[base + 2 * (size_t)N] = tv;    // branch t=2
            }
        }
    }
}

// One block per row: squared L2 norm over N columns (monotonic w.r.t. norm;
// +/- branch pairs remain bit-exact ties).
__global__ __launch_bounds__(256)
void row_sumsq_kernel(const float* __restrict__ A, float* __restrict__ norms, int N) {
    __shared__ float red[256];
    const size_t row = blockIdx.x;
    float s = 0.0f;
    for (int i = threadIdx.x; i < N; i += 256) {
        float v = A[row * (size_t)N + i];
        s += v * v;
    }
    red[threadIdx.x] = s;
    __syncthreads();
    for (int off = 128; off > 0; off >>= 1) {
        if ((int)threadIdx.x < off) red[threadIdx.x] += red[threadIdx.x + off];
        __syncthreads();
    }
    if (threadIdx.x == 0) norms[row] = red[0];
}

// Exact top-`keep` by rank counting. Descending value, ties -> lower index
// (jax.lax.top_k ordering). One block per batch; Pp <= 384.
__global__ __launch_bounds__(256)
void topk_kernel(const float* __restrict__ norms, int* __restrict__ kidx,
                 int Pp, int keep) {
    __shared__ float sn[512];
    const int b = blockIdx.x;
    for (int i = threadIdx.x; i < Pp; i += 256) sn[i] = norms[(size_t)b * Pp + i];
    __syncthreads();
    for (int i = threadIdx.x; i < Pp; i += 256) {
        const float v = sn[i];
        int rank = 0;
        for (int j = 0; j < Pp; ++j) {
            const float u = sn[j];
            rank += (u > v) || (u == v && j < i);
        }
        if (rank < keep) kidx[(size_t)b * keep + rank] = i;
    }
}

// Compact kept rows: out[(b*keep+j), :] = in[(b*Pp + kidx[b][j]), :]
__global__ __launch_bounds__(256)
void gather_rows_kernel(const float* __restrict__ in, const int* __restrict__ kidx,
                        float* __restrict__ out, int Pp, int keep, int N, size_t total) {
    const size_t tid = (size_t)blockIdx.x * blockDim.x + threadIdx.x;
    if (tid >= total) return;
    const int    o  = (int)(tid % (size_t)N);
    const size_t rj = tid / (size_t)N;
    const int    j  = (int)(rj % (size_t)keep);
    const int    b  = (int)(rj / (size_t)keep);
    const int    p  = kidx[(size_t)b * keep + j];
    out[tid] = in[((size_t)b * Pp + p) * (size_t)N + o];
}

// Final gather + transpose into reference layout (B, O=512, P=128):
// dataout[b, o, j] = in[(b*384 + kidx[b][j]), o]
__global__ __launch_bounds__(256)
void final_gather_kernel(const float* __restrict__ in, const int* __restrict__ kidx,
                         float* __restrict__ dataout) {
    const size_t tid = (size_t)blockIdx.x * blockDim.x + threadIdx.x;  // < 128*512*128
    const int j = (int)(tid & 127);
    const int o = (int)((tid >> 7) & 511);
    const int b = (int)(tid >> 16);
    const int p = kidx[b * 128 + j];
    dataout[tid] = in[((size_t)b * 384 + p) * 512 + o];
}

// output[b, o] = data[b, o, argmax_j |data[b,o,j]|]  (first max wins, as argmax)
__global__ __launch_bounds__(256)
void argmax_kernel(const float* __restrict__ data, float* __restrict__ outv) {
    const int tid = blockIdx.x * blockDim.x + threadIdx.x;  // b*512 + o, < 65536
    const float* p = data + (size_t)tid * 128;
    float val  = p[0];
    float best = fabsf(val);
    for (int j = 1; j < 128; ++j) {
        const float v = p[j];
        const float a = fabsf(v);
        if (a > best) { best = a; val = v; }
    }
    outv[tid] = val;
}

// ---------------------------------------------------------------------------
// Host-side orchestration
// ---------------------------------------------------------------------------
static void launch_layer(const float* A, const float* W, const float* b, float* out,
                         int M, int N, int K, bool do_tanh, hipStream_t stream) {
    const int waves   = (M / 64) * (N / 64);   // 64x64 C block per wave
    const int threads = 256;
    const int blocks  = (waves * 32 + threads - 1) / threads;
    if (do_tanh)
        layer_wmma_kernel<true><<<blocks, threads, 0, stream>>>(A, W, b, out, M, N, K);
    else
        layer_wmma_kernel<false><<<blocks, threads, 0, stream>>>(A, W, b, out, M, N, K);
}

extern "C" void kernel_launch(void* const* d_in, const int* in_sizes, int n_in,
                              void* d_out, int out_size, void* d_ws, size_t ws_size,
                              hipStream_t stream) {
    const float* x = (const float*)d_in[0];
    const float* Wt[6];
    const float* bt[6];
    for (int i = 0; i < 6; ++i) {
        Wt[i] = (const float*)d_in[1 + 2 * i];
        bt[i] = (const float*)d_in[2 + 2 * i];
    }

    // Workspace layout (floats):
    //   buf0: 32M floats (max 31104x1024 layer-4 output, 49152x512 layer-5 output)
    //   buf1: 18M floats (max 16384x1024 pruned activations)
    //   norms: 49152 floats; kidx: 128*128 ints
    float* ws    = (float*)d_ws;
    float* buf0  = ws;
    float* buf1  = ws + ((size_t)32u << 20);
    float* norms = ws + ((size_t)50u << 20);
    int*   kidx  = (int*)(norms + 65536);

    // Layers 0..4 (tanh), paths 1 -> 3 -> 9 -> 27 -> 81 -> 243
    launch_layer(x,    Wt[0], bt[0], buf0,   128, 1024, 1024, true, stream);  // out rows 384
    launch_layer(buf0, Wt[1], bt[1], buf1,   384, 1024, 1024, true, stream);  // 1152
    launch_layer(buf1, Wt[2], bt[2], buf0,  1152, 1024, 1024, true, stream);  // 3456
    launch_layer(buf0, Wt[3], bt[3], buf1,  3456, 1024, 1024, true, stream);  // 10368
    launch_layer(buf1, Wt[4], bt[4], buf0, 10368, 1024, 1024, true, stream);  // 31104

    // Prune 243 -> 128 per batch
    row_sumsq_kernel<<<31104, 256, 0, stream>>>(buf0, norms, 1024);
    topk_kernel<<<128, 256, 0, stream>>>(norms, kidx, 243, 128);
    {
        const size_t total = (size_t)128 * 128 * 1024;
        gather_rows_kernel<<<(int)((total + 255) / 256), 256, 0, stream>>>(
            buf0, kidx, buf1, 243, 128, 1024, total);
    }

    // Layer 5 (no tanh), paths 128 -> 384, N = 512
    launch_layer(buf1, Wt[5], bt[5], buf0, 16384, 512, 1024, false, stream);  // 49152 rows

    // Prune 384 -> 128 per batch
    row_sumsq_kernel<<<49152, 256, 0, stream>>>(buf0, norms, 512);
    topk_kernel<<<128, 256, 0, stream>>>(norms, kidx, 384, 128);

    // d_out: [output (128*512) | data (128*512*128)]
    float* out_vals = (float*)d_out;
    float* data_out = out_vals + 128 * 512;
    final_gather_kernel<<<(128 * 512 * 128) / 256, 256, 0, stream>>>(buf0, kidx, data_out);
    argmax_kernel<<<(128 * 512) / 256, 256, 0, stream>>>(data_out, out_vals);
}